// TopKRouter_56367150793178
// MI455X (gfx1250) — compile-verified
//
#include <hip/hip_runtime.h>
#include <hip/hip_bf16.h>
#include <math.h>

typedef __attribute__((ext_vector_type(2))) float v2f;
typedef __attribute__((ext_vector_type(8))) float v8f;

#define HDIM 2048
#define NEXP 64
#define TOPK 8
#define WAVES_PER_BLOCK 4
#define TOK_PER_BLOCK (WAVES_PER_BLOCK * 16)
#define LDS_STRIDE 65   // 64 experts + 1 pad -> conflict-free row reads

__launch_bounds__(128, 1)
__global__ void topk_router_fused(const float* __restrict__ hidden,
                                  const float* __restrict__ gate_w,
                                  const float* __restrict__ expert_bias,
                                  float* __restrict__ out_probs,
                                  float* __restrict__ out_idx,
                                  float* __restrict__ out_map,
                                  float* __restrict__ out_aux,
                                  int T) {
    __shared__ float lds[TOK_PER_BLOCK * LDS_STRIDE];
    __shared__ float lbias[NEXP];

    const int tid   = threadIdx.x;
    const int wave  = tid >> 5;
    const int lane  = tid & 31;
    const int row   = lane & 15;          // M row (A) / N col (B) within 16-tile
    const int khalf = (lane >> 4) << 1;   // lanes 0-15 -> K{0,1}, lanes 16-31 -> K{2,3}

    const int tokenBase = blockIdx.x * TOK_PER_BLOCK + wave * 16;

    // --- Phase 1: logits tile [16 tokens x 64 experts] via f32 WMMA ---
    const float* aptr  = hidden + (size_t)(tokenBase + row) * HDIM + khalf;
    const float* bptr0 = gate_w + (size_t)( 0 + row) * HDIM + khalf;
    const float* bptr1 = gate_w + (size_t)(16 + row) * HDIM + khalf;
    const float* bptr2 = gate_w + (size_t)(32 + row) * HDIM + khalf;
    const float* bptr3 = gate_w + (size_t)(48 + row) * HDIM + khalf;

    v8f acc0 = {}; v8f acc1 = {}; v8f acc2 = {}; v8f acc3 = {};

    #pragma unroll 8
    for (int k = 0; k < HDIM; k += 4) {
        v2f a  = *(const v2f*)(aptr  + k);
        v2f b0 = *(const v2f*)(bptr0 + k);
        v2f b1 = *(const v2f*)(bptr1 + k);
        v2f b2 = *(const v2f*)(bptr2 + k);
        v2f b3 = *(const v2f*)(bptr3 + k);
        acc0 = __builtin_amdgcn_wmma_f32_16x16x4_f32(false, a, false, b0, (short)0, acc0, false, false);
        acc1 = __builtin_amdgcn_wmma_f32_16x16x4_f32(false, a, false, b1, (short)0, acc1, false, false);
        acc2 = __builtin_amdgcn_wmma_f32_16x16x4_f32(false, a, false, b2, (short)0, acc2, false, false);
        acc3 = __builtin_amdgcn_wmma_f32_16x16x4_f32(false, a, false, b3, (short)0, acc3, false, false);
    }

    // Scatter C/D layout (VGPR v: M=v lanes 0-15, M=v+8 lanes 16-31; N=lane&15) to LDS
    const int tlBase = wave * 16 + ((lane >> 4) * 8);
    #pragma unroll
    for (int v = 0; v < 8; ++v) {
        const int tl = tlBase + v;
        lds[tl * LDS_STRIDE +  0 + row] = acc0[v];
        lds[tl * LDS_STRIDE + 16 + row] = acc1[v];
        lds[tl * LDS_STRIDE + 32 + row] = acc2[v];
        lds[tl * LDS_STRIDE + 48 + row] = acc3[v];
    }
    if (tid < NEXP) lbias[tid] = expert_bias[tid];
    __syncthreads();

    // --- Phase 2: one thread per token: softmax + biased top-8 + renorm ---
    if (tid < TOK_PER_BLOCK) {
        const int token = blockIdx.x * TOK_PER_BLOCK + tid;
        float* lrow = &lds[tid * LDS_STRIDE];

        float m = -INFINITY;
        for (int e = 0; e < NEXP; ++e) m = fmaxf(m, lrow[e]);

        float ssum = 0.0f;
        for (int e = 0; e < NEXP; ++e) {
            float v = __expf(lrow[e] - m);
            lrow[e] = v;                 // overwrite logits with unnormalized exp
            ssum += v;
        }
        const float invs = 1.0f / ssum;  // score[e] = lrow[e] * invs

        unsigned long long taken = 0ull;
        float tops[TOPK];
        int   ids[TOPK];
        float tsum = 0.0f;

        #pragma unroll
        for (int k = 0; k < TOPK; ++k) {
            float best = -INFINITY;
            int   bid  = 0;
            for (int e = 0; e < NEXP; ++e) {
                const float sel = lrow[e] * invs + lbias[e];
                const bool free = ((taken >> e) & 1ull) == 0ull;
                if (free && sel > best) { best = sel; bid = e; }
            }
            taken |= (1ull << bid);
            const float sc = lrow[bid] * invs;   // original softmax prob
            tops[k] = sc;
            ids[k]  = bid;
            tsum   += sc;
        }
        const float rnorm = 1.0f / (tsum + 1e-9f);

        #pragma unroll
        for (int k = 0; k < TOPK; ++k) {
            out_probs[(size_t)token * TOPK + k] = tops[k] * rnorm;
            out_idx  [(size_t)token * TOPK + k] = (float)ids[k];
        }
        for (int e = 0; e < NEXP; ++e) {
            out_map[(size_t)token * NEXP + e] = ((taken >> e) & 1ull) ? 1.0f : 0.0f;
        }
    }
    if (blockIdx.x == 0 && tid == 0) out_aux[0] = 0.0f;
}

extern "C" void kernel_launch(void* const* d_in, const int* in_sizes, int n_in,
                              void* d_out, int out_size, void* d_ws, size_t ws_size,
                              hipStream_t stream) {
    const float* hidden      = (const float*)d_in[0];   // [T, H]
    const float* gate_w      = (const float*)d_in[1];   // [E, H]
    const float* expert_bias = (const float*)d_in[2];   // [E]
    const int T = in_sizes[0] / HDIM;

    float* out   = (float*)d_out;
    float* probs = out;                                        // [T, 8]
    float* idx   = out + (size_t)T * TOPK;                     // [T, 8]
    float* map   = out + (size_t)2 * T * TOPK;                 // [T, 64]
    float* aux   = out + (size_t)2 * T * TOPK + (size_t)T * NEXP; // scalar

    const int blocks = (T + TOK_PER_BLOCK - 1) / TOK_PER_BLOCK;
    topk_router_fused<<<blocks, WAVES_PER_BLOCK * 32, 0, stream>>>(
        hidden, gate_w, expert_bias, probs, idx, map, aux, T);
}